// HungarianMatcher_66477503807569
// MI455X (gfx1250) — compile-verified
//
#include <hip/hip_runtime.h>
#include <stdint.h>

// Problem constants (match reference)
#define BSZ 16
#define QN  1500
#define KC  256
#define NTT 150
#define TT  (BSZ*NTT)   // 2400 targets
#define NN  (BSZ*QN)    // 24000 predictions
#define RB  32          // prediction rows per block in main kernel

__device__ __forceinline__ float frcp(float x) { return __builtin_amdgcn_rcpf(x); }

// Raw hardware transcendentals (inputs are always normal floats here, so the
// compiler's denormal-rescue sequences around __logf/__expf are dead weight).
__device__ __forceinline__ float fast_log(float x) {        // ln(x), x >= 1e-8
    return __builtin_amdgcn_logf(x) * 0.6931471805599453f;  // v_log_f32 is log2
}
__device__ __forceinline__ float fast_exp(float x) {        // e^x
    return __builtin_amdgcn_exp2f(x * 1.4426950408889634f); // v_exp_f32 is exp2
}

// ---------------------------------------------------------------------------
// Kernel 1: focal classification cost table  [N, K]
//   table[n,k] = ALPHA*(1-p)^2*(-log(p+EPS)) - (1-ALPHA)*p^2*(-log(1-p+EPS))
// ---------------------------------------------------------------------------
__global__ __launch_bounds__(256) void prep_class(const float* __restrict__ logits,
                                                  float* __restrict__ table) {
    int i = blockIdx.x * 256 + threadIdx.x;
    if (i >= NN * KC) return;
    float x   = logits[i];
    float p   = frcp(1.0f + fast_exp(-x));    // sigmoid
    float omp = 1.0f - p;
    float pos = 0.25f * omp * omp * (-fast_log(p   + 1e-8f));
    float neg = 0.75f * p   * p   * (-fast_log(omp + 1e-8f));
    table[i] = pos - neg;
}

// ---------------------------------------------------------------------------
// Kernel 2: box preprocessing.
//   predD: AoS stride 12 (16B aligned rows): cx,cy,w,h,x0,y0,x1,y1,area
//   tgtD : SoA [9][T] of the same 9 values (coalesced column access)
// ---------------------------------------------------------------------------
__global__ __launch_bounds__(256) void prep_boxes(const float* __restrict__ pb,
                                                  const float* __restrict__ tb,
                                                  float* __restrict__ predD,
                                                  float* __restrict__ tgtD) {
    int i = blockIdx.x * 256 + threadIdx.x;
    if (i < NN) {
        float cx = pb[i*4+0], cy = pb[i*4+1], w = pb[i*4+2], h = pb[i*4+3];
        float x0 = cx - 0.5f*w, y0 = cy - 0.5f*h;
        float x1 = cx + 0.5f*w, y1 = cy + 0.5f*h;
        float area = (x1 - x0) * (y1 - y0);
        float* o = predD + (size_t)i * 12;
        o[0]=cx; o[1]=cy; o[2]=w; o[3]=h; o[4]=x0; o[5]=y0; o[6]=x1; o[7]=y1; o[8]=area;
        o[9]=0.f; o[10]=0.f; o[11]=0.f;
    }
    if (i < TT) {
        float cx = tb[i*4+0], cy = tb[i*4+1], w = tb[i*4+2], h = tb[i*4+3];
        float x0 = cx - 0.5f*w, y0 = cy - 0.5f*h;
        float x1 = cx + 0.5f*w, y1 = cy + 0.5f*h;
        float area = (x1 - x0) * (y1 - y0);
        tgtD[0*TT+i]=cx; tgtD[1*TT+i]=cy; tgtD[2*TT+i]=w;  tgtD[3*TT+i]=h;
        tgtD[4*TT+i]=x0; tgtD[5*TT+i]=y0; tgtD[6*TT+i]=x1; tgtD[7*TT+i]=y1;
        tgtD[8*TT+i]=area;
    }
}

// ---------------------------------------------------------------------------
// Kernel 3: main pairwise cost.  Block = 256 threads, owns RB=32 rows x T cols.
//   - 32KB focal-table slab staged to LDS via CDNA5 async global->LDS DMA
//     (global_load_async_to_lds_b128, tracked by ASYNCcnt)
//   - target data held in registers across the 32-row inner loop
//   - pred-row data is wave-uniform -> scalar loads
//   - non-temporal stores for the 230MB write-once output (keep L2 for the
//     hot target set + streamed focal table)
// ---------------------------------------------------------------------------
__global__ __launch_bounds__(256) void matcher_main(const float* __restrict__ table,
                                                    const float* __restrict__ predD,
                                                    const float* __restrict__ tgtD,
                                                    const int*   __restrict__ tgt_ids,
                                                    float* __restrict__ out) {
    __shared__ __align__(16) float sCls[RB * KC];   // 32 KB
    const int tid = threadIdx.x;
    const int r0  = blockIdx.x * RB;

    // --- async-stage class-cost rows [r0, r0+RB) into LDS (ASYNCcnt path) ---
    {
        const float* gbase = table + (size_t)r0 * KC;        // 16B aligned
        uint32_t lbase = (uint32_t)(uintptr_t)(&sCls[0]);    // LDS byte offset (low 32 bits)
        #pragma unroll
        for (int i = 0; i < (RB * KC * 4) / (256 * 16); ++i) {   // 8 x b128 per lane
            uint32_t off   = (uint32_t)(tid * 16 + i * 256 * 16);
            uint32_t laddr = lbase + off;
            asm volatile("global_load_async_to_lds_b128 %0, %1, %2"
                         :: "v"(laddr), "v"(off), "s"(gbase) : "memory");
        }
        asm volatile("s_wait_asynccnt 0" ::: "memory");
    }
    __syncthreads();

    // --- column chunks: each lane keeps its target's 10 values in registers ---
    for (int c = 0; c < (TT + 255) / 256; ++c) {
        int  col   = c * 256 + tid;
        bool valid = col < TT;
        int  cc    = valid ? col : (TT - 1);      // clamp to keep loads in-bounds
        float tcx = tgtD[0*TT+cc], tcy = tgtD[1*TT+cc];
        float tw  = tgtD[2*TT+cc], th  = tgtD[3*TT+cc];
        float tx0 = tgtD[4*TT+cc], ty0 = tgtD[5*TT+cc];
        float tx1 = tgtD[6*TT+cc], ty1 = tgtD[7*TT+cc];
        float tarea = tgtD[8*TT+cc];
        int   id  = tgt_ids[cc];

        #pragma unroll 4
        for (int r = 0; r < RB; ++r) {
            const float* pr = predD + (size_t)(r0 + r) * 12;   // uniform -> s_load
            float pcx = pr[0], pcy = pr[1], pw = pr[2], ph = pr[3];
            float px0 = pr[4], py0 = pr[5], px1 = pr[6], py1 = pr[7];
            float parea = pr[8];

            // L1 in cxcywh space
            float l1 = fabsf(pcx - tcx) + fabsf(pcy - tcy)
                     + fabsf(pw  - tw ) + fabsf(ph  - th );

            // GIoU in xyxy space
            float ix0 = fmaxf(px0, tx0), iy0 = fmaxf(py0, ty0);
            float ix1 = fminf(px1, tx1), iy1 = fminf(py1, ty1);
            float iw  = fmaxf(ix1 - ix0, 0.0f), ih = fmaxf(iy1 - iy0, 0.0f);
            float inter = iw * ih;
            float uni   = parea + tarea - inter;
            float ex0 = fminf(px0, tx0), ey0 = fminf(py0, ty0);
            float ex1 = fmaxf(px1, tx1), ey1 = fmaxf(py1, ty1);
            float earea = (ex1 - ex0) * (ey1 - ey0);   // always >= 0 (max>=min)
            float iou   = inter * frcp(uni);
            float giou  = iou - (earea - uni) * frcp(earea);

            // class cost: single LDS gather from the staged focal-table slab
            float ccost = sCls[r * KC + id];

            float Cv = fmaf(-2.0f, giou, fmaf(5.0f, l1, ccost));
            if (valid)
                __builtin_nontemporal_store(Cv, &out[(size_t)(r0 + r) * TT + col]);
        }
    }
}

// ---------------------------------------------------------------------------
// Host launcher
// ---------------------------------------------------------------------------
extern "C" void kernel_launch(void* const* d_in, const int* in_sizes, int n_in,
                              void* d_out, int out_size, void* d_ws, size_t ws_size,
                              hipStream_t stream) {
    const float* pred_logits = (const float*)d_in[0];   // [16,1500,256]
    const float* pred_boxes  = (const float*)d_in[1];   // [16,1500,4]
    const float* tgt_bbox    = (const float*)d_in[2];   // [2400,4]
    const int*   tgt_ids     = (const int*)  d_in[3];   // [2400]
    float*       out         = (float*)d_out;           // [16,1500,2400]

    // Workspace layout (all 16B aligned); total ~25.9 MB
    float* table = (float*)d_ws;                 // N*K     = 6,144,000 f  (24.6 MB)
    float* predD = table + (size_t)NN * KC;      // N*12    =   288,000 f
    float* tgtD  = predD + (size_t)NN * 12;      // 9*T     =    21,600 f

    prep_class <<<(NN * KC + 255) / 256, 256, 0, stream>>>(pred_logits, table);
    prep_boxes <<<(NN + 255) / 256,      256, 0, stream>>>(pred_boxes, tgt_bbox, predD, tgtD);
    matcher_main<<<NN / RB,              256, 0, stream>>>(table, predD, tgtD, tgt_ids, out);
}